// GroupedQueryAttentionCache__64287070486907
// MI455X (gfx1250) — compile-verified
//
#include <hip/hip_runtime.h>
#include <hip/hip_bf16.h>
#include <stdint.h>

// ---------------------------------------------------------------------------
// GroupedQueryAttentionCache append (offset != 0 path):
//   out_k[b, 0:8192] = k_cache[b];  out_k[b, 8192] = k[b]
//   out_v[b, 0:8192] = v_cache[b];  out_v[b, 8192] = v[b]
// Pure bandwidth problem (~512 MiB traffic, 0 FLOPs) -> use the CDNA5
// Tensor Data Mover (async DMA, TENSORcnt) for the bulk 256 MiB cache copy.
// ---------------------------------------------------------------------------

typedef unsigned int v4u __attribute__((ext_vector_type(4)));
typedef int          v8i __attribute__((ext_vector_type(8)));
typedef int          v4i __attribute__((ext_vector_type(4)));

// Geometry (elements are bf16 = 2 bytes)
constexpr uint32_t Bn      = 8;
constexpr uint32_t S_CACHE = 8192;
constexpr uint32_t ROW     = 8 * 128;                                 // H_kv * D = 1024
constexpr uint64_t CACHE_ELEMS_PER_B = (uint64_t)S_CACHE * ROW;       // 8,388,608
constexpr uint64_t OUT_ELEMS_PER_B   = (uint64_t)(S_CACHE + 1) * ROW; // 8,389,632
constexpr uint64_t OUT_ELEMS_PER_T   = (uint64_t)Bn * OUT_ELEMS_PER_B;// 67,117,056

constexpr uint32_t TILE_BYTES      = 65536;                           // 64 KiB LDS tile
constexpr uint32_t TILES_PER_CHUNK = (uint32_t)((CACHE_ELEMS_PER_B * 2) / TILE_BYTES); // 256
constexpr uint32_t NUM_CHUNKS      = 16;                              // {k,v} x 8 batches
constexpr uint32_t NUM_TILES       = NUM_CHUNKS * TILES_PER_CHUNK;    // 4096

// Detect TDM builtin availability (device pass only; host pass parses fallback).
#if defined(__AMDGCN__)
#  if __has_builtin(__builtin_amdgcn_tensor_load_to_lds) && \
      __has_builtin(__builtin_amdgcn_tensor_store_from_lds)
#    define ATH_HAS_TDM 1
#  endif
#endif
#ifndef ATH_HAS_TDM
#  define ATH_HAS_TDM 0
#endif

// ---------------------------------------------------------------------------
// Bulk cache copy: one 64 KiB tile per workgroup (1 wave32), via TDM.
// D# descriptor (CDNA5 ISA):
//   group0: {count=1, lds_addr, global_addr[56:0], type=2}
//   group1: data_size=8B, tensor_dim0=tile_dim0=stride=8192 units, dim1=1
//   group2/3: zero (<=2D tensor)
// All descriptor math is uniform (blockIdx-derived) -> stays in SALU/SGPRs.
// ---------------------------------------------------------------------------
__global__ __launch_bounds__(32) void kvcache_bulk_copy(
    const __hip_bfloat16* __restrict__ kc,
    const __hip_bfloat16* __restrict__ vc,
    __hip_bfloat16* __restrict__ out)
{
    const uint32_t tile  = blockIdx.x;                 // 0 .. NUM_TILES-1 (uniform)
    const uint32_t chunk = tile / TILES_PER_CHUNK;     // 0 .. 15
    const uint32_t tic   = tile % TILES_PER_CHUNK;     // tile-in-chunk
    const uint32_t t     = chunk >> 3;                 // 0 = k, 1 = v
    const uint32_t b     = chunk & 7;                  // batch

    const __hip_bfloat16* src_base = t ? vc : kc;
    const uint64_t src_addr = (uint64_t)(uintptr_t)(src_base + b * CACHE_ELEMS_PER_B)
                              + (uint64_t)tic * TILE_BYTES;
    const uint64_t dst_addr = (uint64_t)(uintptr_t)(out + t * OUT_ELEMS_PER_T
                                                        + b * OUT_ELEMS_PER_B)
                              + (uint64_t)tic * TILE_BYTES;

#if ATH_HAS_TDM
    __shared__ alignas(128) unsigned char smem[TILE_BYTES];
    const uint32_t lds_off = (uint32_t)(uintptr_t)(&smem[0]); // low 32b of generic LDS ptr = LDS byte offset

    const uint32_t UNITS = TILE_BYTES / 8;             // 8192 units of data_size=8B

    // D# group 1 (shared by load and store)
    v8i g1;
    g1[0] = (int)(3u << 16);                           // workgroup_mask=0, data_size=3 (8B), no flags
    g1[1] = (int)((UNITS & 0xFFFFu) << 16);            // atomic_barrier_addr=0 | tensor_dim0[15:0]
    g1[2] = (int)((UNITS >> 16) | (1u << 16));         // tensor_dim0[31:16] | tensor_dim1[15:0]=1
    g1[3] = (int)((UNITS & 0xFFFFu) << 16);            // tensor_dim1[31:16]=0 | tile_dim0
    g1[4] = 1;                                         // tile_dim1=1 | tile_dim2=0
    g1[5] = (int)UNITS;                                // tensor_dim0_stride[31:0]
    g1[6] = 0;                                         // stride0[47:32] | stride1[15:0]
    g1[7] = 0;                                         // stride1[47:16]

    v4u g0l = { 1u, lds_off, (uint32_t)src_addr,
                (uint32_t)((src_addr >> 32) & 0x1FFFFFFu) | (2u << 30) };
    v4u g0s = { 1u, lds_off, (uint32_t)dst_addr,
                (uint32_t)((dst_addr >> 32) & 0x1FFFFFFu) | (2u << 30) };
    v4i z4 = { 0, 0, 0, 0 };

#if __has_include(<hip/amd_detail/amd_gfx1250_TDM.h>)
    // therock-10.0 header toolchain: 6-arg builtin form
    v8i z8 = { 0, 0, 0, 0, 0, 0, 0, 0 };
    __builtin_amdgcn_tensor_load_to_lds(g0l, g1, z4, z4, z8, 0);
    __builtin_amdgcn_s_wait_tensorcnt(0);
    __builtin_amdgcn_tensor_store_from_lds(g0s, g1, z4, z4, z8, 0);
#else
    // ROCm 7.2 (clang-22): 5-arg builtin form
    __builtin_amdgcn_tensor_load_to_lds(g0l, g1, z4, z4, 0);
    __builtin_amdgcn_s_wait_tensorcnt(0);
    __builtin_amdgcn_tensor_store_from_lds(g0s, g1, z4, z4, 0);
#endif
    __builtin_amdgcn_s_wait_tensorcnt(0);
#else
    // Fallback (host parse / toolchains without TDM builtins):
    // direct streaming copy, 16 B per lane per iteration, nontemporal.
    // Uses clang ext_vector (v4u), which __builtin_nontemporal_* accepts.
    const v4u* __restrict__ s = (const v4u*)(uintptr_t)src_addr;
    v4u* __restrict__       d = (v4u*)(uintptr_t)dst_addr;
    for (uint32_t i = threadIdx.x; i < TILE_BYTES / 16; i += 32) {
        __builtin_nontemporal_store(__builtin_nontemporal_load(s + i), d + i);
    }
#endif
}

// ---------------------------------------------------------------------------
// Appended rows: out_t[b, 8192, :, :] = new_t[b]  (2 tensors x 8 batches x 2 KiB)
// 2048 threads, one 16-byte vector (8 bf16) each -> global_load/store_b128.
// ---------------------------------------------------------------------------
__global__ void kvcache_tail_copy(
    const __hip_bfloat16* __restrict__ knew,
    const __hip_bfloat16* __restrict__ vnew,
    __hip_bfloat16* __restrict__ out)
{
    const uint32_t i = blockIdx.x * blockDim.x + threadIdx.x; // 0 .. 2047
    const uint32_t t = i >> 10;        // tensor select (0=k, 1=v)
    const uint32_t j = i & 1023;       // 16B-vector index within [8 batches x 128]
    const uint32_t b = j >> 7;
    const uint32_t r = j & 127;

    const v4u* __restrict__ src = (const v4u*)(t ? vnew : knew);
    v4u* __restrict__ dst = (v4u*)out;
    const uint64_t dst4 = (uint64_t)t * (OUT_ELEMS_PER_T / 8)
                        + (uint64_t)b * (OUT_ELEMS_PER_B / 8)
                        + (CACHE_ELEMS_PER_B / 8)
                        + r;
    dst[dst4] = src[j];
}

extern "C" void kernel_launch(void* const* d_in, const int* in_sizes, int n_in,
                              void* d_out, int out_size, void* d_ws, size_t ws_size,
                              hipStream_t stream)
{
    (void)in_sizes; (void)n_in; (void)out_size; (void)d_ws; (void)ws_size;
    const __hip_bfloat16* kc = (const __hip_bfloat16*)d_in[0];
    const __hip_bfloat16* vc = (const __hip_bfloat16*)d_in[1];
    const __hip_bfloat16* k  = (const __hip_bfloat16*)d_in[2];
    const __hip_bfloat16* v  = (const __hip_bfloat16*)d_in[3];
    // d_in[4] = offset (static 8192 in this harness; offset==0 branch resolves
    // at trace time in the reference, so it is not a runtime input here).
    __hip_bfloat16* out = (__hip_bfloat16*)d_out;

    kvcache_bulk_copy<<<NUM_TILES, 32, 0, stream>>>(kc, vc, out);
    kvcache_tail_copy<<<8, 256, 0, stream>>>(k, v, out);
}